// MPS_33930241638883
// MI455X (gfx1250) — compile-verified
//
#include <hip/hip_runtime.h>
#include <math.h>

// ---------------------------------------------------------------------------
// 20-qubit statevector simulator.
// psi (float2[2^20]) lives in d_ws, the 8x8 realified gate matrix M follows it.
// ---------------------------------------------------------------------------

typedef float v2f __attribute__((ext_vector_type(2)));
typedef float v8f __attribute__((ext_vector_type(8)));

#define NSTATE  (1 << 20)
#define NGROUP  (1 << 18)
#define TPB     256
#define WPB     (TPB / 32)
#define RSTR    10            // LDS row stride in floats (40B: 8B-aligned, conflict-free)

// ------------------------- complex helpers (prep kernel) -------------------
struct C64 { float re, im; };
__device__ inline C64 cmul(C64 a, C64 b) {
    return { a.re * b.re - a.im * b.im, a.re * b.im + a.im * b.re };
}
__device__ inline C64 cadd(C64 a, C64 b) { return { a.re + b.re, a.im + b.im }; }

__device__ void mat4_mul(const C64* A, const C64* B, C64* out) {
    for (int i = 0; i < 4; ++i)
        for (int j = 0; j < 4; ++j) {
            C64 s = {0.f, 0.f};
            for (int k = 0; k < 4; ++k) s = cadd(s, cmul(A[i*4+k], B[k*4+j]));
            out[i*4+j] = s;
        }
}
__device__ void kron2(const C64* A, const C64* B, C64* out) {
    // first factor acts on wire q (high bit a), second on wire q+1 (low bit b)
    for (int i = 0; i < 2; ++i)
        for (int k = 0; k < 2; ++k)
            for (int j = 0; j < 2; ++j)
                for (int l = 0; l < 2; ++l)
                    out[(i*2+k)*4 + (j*2+l)] = cmul(A[i*2+j], B[k*2+l]);
}
__device__ void mat2_id(C64* A) {
    A[0] = {1.f,0.f}; A[1] = {0.f,0.f}; A[2] = {0.f,0.f}; A[3] = {1.f,0.f};
}
// rotation for the 'I'-containing Pauli strings: RX/RY/RZ(theta)
__device__ void rot_mat(int p, float th, C64* R) {
    float h = 0.5f * th, c = cosf(h), s = sinf(h);
    if (p == 1)      { R[0]={c,0.f};  R[1]={0.f,-s}; R[2]={0.f,-s}; R[3]={c,0.f}; } // RX
    else if (p == 2) { R[0]={c,0.f};  R[1]={-s,0.f}; R[2]={s,0.f};  R[3]={c,0.f}; } // RY
    else             { R[0]={c,-s};   R[1]={0.f,0.f};R[2]={0.f,0.f};R[3]={c, s};  } // RZ
}
// basis change for the two-Pauli strings: X -> RY(-pi/2), Y -> RX(+pi/2), Z -> I
__device__ void basis_mat(int p, C64* A) {
    const float c = 0.70710678118654752f;
    if (p == 1)      { A[0]={c,0.f}; A[1]={c,0.f};   A[2]={-c,0.f};  A[3]={c,0.f}; }
    else if (p == 2) { A[0]={c,0.f}; A[1]={0.f,-c};  A[2]={0.f,-c};  A[3]={c,0.f}; }
    else             { mat2_id(A); }
}

__global__ void build_matrix_kernel(const float* __restrict__ params,
                                    float* __restrict__ Mout) {
    if (threadIdx.x != 0 || blockIdx.x != 0) return;

    const int P0[15] = {0,0,0, 1,1,1,1, 2,2,2,2, 3,3,3,3};
    const int P1[15] = {1,2,3, 0,1,2,3, 0,1,2,3, 0,1,2,3};

    C64 I2[4]; mat2_id(I2);
    C64 CN[16];
    for (int i = 0; i < 16; ++i) CN[i] = {0.f, 0.f};
    CN[0*4+0] = {1.f,0.f}; CN[1*4+1] = {1.f,0.f};   // CNOT(control=a, target=b)
    CN[2*4+3] = {1.f,0.f}; CN[3*4+2] = {1.f,0.f};

    C64 U[16];
    for (int i = 0; i < 16; ++i) U[i] = {0.f, 0.f};
    U[0] = U[5] = U[10] = U[15] = {1.f, 0.f};

    for (int i = 0; i < 15; ++i) {
        float th = params[i];
        int p0 = P0[i], p1 = P1[i];
        C64 G[16];
        if (p0 == 0) {                 // 'I?' : rotate wire q+1
            C64 R[4]; rot_mat(p1, th, R); kron2(I2, R, G);
        } else if (p1 == 0) {          // '?I' : rotate wire q
            C64 R[4]; rot_mat(p0, th, R); kron2(R, I2, G);
        } else {                       // Pre . CNOT . RZ_b . CNOT . Pre (faithful)
            C64 A0[4], A1[4], Pre[16], RZb[16], R[4], T1[16], T2[16];
            basis_mat(p0, A0); basis_mat(p1, A1);
            kron2(A0, A1, Pre);
            rot_mat(3, th, R); kron2(I2, R, RZb);
            mat4_mul(CN, Pre, T1);
            mat4_mul(RZb, T1, T2);
            mat4_mul(CN, T2, T1);
            mat4_mul(Pre, T1, G);
        }
        C64 T[16]; mat4_mul(G, U, T);
        for (int k = 0; k < 16; ++k) U[k] = T[k];
    }

    // realify: component vector c = [re0,im0,re1,im1,re2,im2,re3,im3]
    // M[2j  ][2k] =  Re U[j][k]   M[2j  ][2k+1] = -Im U[j][k]
    // M[2j+1][2k] =  Im U[j][k]   M[2j+1][2k+1] =  Re U[j][k]
    for (int j = 0; j < 4; ++j)
        for (int k = 0; k < 4; ++k) {
            C64 u = U[j*4+k];
            Mout[(2*j  )*8 + 2*k    ] =  u.re;
            Mout[(2*j  )*8 + 2*k + 1] = -u.im;
            Mout[(2*j+1)*8 + 2*k    ] =  u.im;
            Mout[(2*j+1)*8 + 2*k + 1] =  u.re;
        }
}

// ------------------------- state init / output -----------------------------
__global__ void init_state_kernel(float2* __restrict__ psi) {
    unsigned i = blockIdx.x * blockDim.x + threadIdx.x;
    float2 v; v.x = (i == 0) ? 1.f : 0.f; v.y = 0.f;
    psi[i] = v;
}

__global__ void probs_kernel(const float2* __restrict__ psi, float* __restrict__ out) {
    unsigned i = blockIdx.x * blockDim.x + threadIdx.x;
    float2 v = psi[i];
    out[i] = v.x * v.x + v.y * v.y;
}

// ------------------------- gate application --------------------------------
// Apply the 4x4 complex gate (as 8x8 real M) to wire pair (q, q+1), in place.
// Each lane owns one group of 4 amplitudes; each wave computes two 16-group
// tiles of out^T(16x8) = in^T(16x8) . M^T(8x8) with chained f32 WMMA.
__global__ void __launch_bounds__(TPB)
apply_gate_kernel(float2* __restrict__ psi, const float* __restrict__ M, int q) {
    __shared__ float lin [WPB][32][RSTR];
    __shared__ float lout[WPB][32][RSTR];

    const int lane = threadIdx.x & 31;
    const int w    = threadIdx.x >> 5;
    const unsigned g = blockIdx.x * TPB + threadIdx.x;   // group id, 0..2^18-1

    const int yb = 18 - q;
    const unsigned ymask = (1u << yb) - 1u;
    const unsigned s     = 1u << yb;                     // stride of bit (q+1)
    const unsigned base  = ((g >> yb) << (yb + 2)) | (g & ymask);

    // gather this group's 4 complex amplitudes -> 8 real components in LDS
    float2 v0 = psi[base];
    float2 v1 = psi[base + s];
    float2 v2 = psi[base + 2*s];
    float2 v3 = psi[base + 3*s];
    float2* row = (float2*)lin[w][lane];                 // 40B rows -> 8B aligned
    row[0] = v0; row[1] = v1; row[2] = v2; row[3] = v3;
    __syncthreads();

    const int hi = lane >> 4;        // 0: K=0/1 (or rows 0..7 of D), 1: K=2/3 (rows 8..15)
    const int n  = lane & 15;        // B column / D column

    // B operands: B[k][n] = M^T[k][n] = M[n][k]; columns 8..15 zero-padded
    v2f b0, b1;
    const float* mr = &M[n * 8 + hi * 2];
    bool nv = (n < 8);
    b0.x = nv ? mr[0] : 0.f;  b0.y = nv ? mr[1] : 0.f;   // K-chunk 0..3
    b1.x = nv ? mr[4] : 0.f;  b1.y = nv ? mr[5] : 0.f;   // K-chunk 4..7

    for (int t = 0; t < 2; ++t) {
        // A (16x4 per chunk): lane L holds row (L&15), K = {0,1}|{2,3} by lane half
        const float* ar = lin[w][t * 16 + n];
        v2f a0; a0.x = ar[hi*2    ]; a0.y = ar[hi*2 + 1];
        v2f a1; a1.x = ar[4 + hi*2]; a1.y = ar[4 + hi*2 + 1];

        v8f d = {};
        d = __builtin_amdgcn_wmma_f32_16x16x4_f32(false, a0, false, b0,
                                                  (short)0, d, false, false);
        d = __builtin_amdgcn_wmma_f32_16x16x4_f32(false, a1, false, b1,
                                                  (short)0, d, false, false);

        // D 16x16: VGPR v holds (M = v + 8*hi, N = n). Scatter to out tile.
        if (n < 8) {
            #pragma unroll
            for (int v = 0; v < 8; ++v)
                lout[w][t * 16 + v + hi * 8][n] = d[v];
        }
    }
    __syncthreads();

    // coalesced write-back of this lane's group
    const float2* orow = (const float2*)lout[w][lane];
    psi[base]       = orow[0];
    psi[base + s]   = orow[1];
    psi[base + 2*s] = orow[2];
    psi[base + 3*s] = orow[3];
}

// ---------------------------------------------------------------------------
extern "C" void kernel_launch(void* const* d_in, const int* in_sizes, int n_in,
                              void* d_out, int out_size, void* d_ws, size_t ws_size,
                              hipStream_t stream) {
    const float* params = (const float*)d_in[0];       // (K=1, 285) fp32; only [0..14] used
    float2* psi = (float2*)d_ws;                       // 8 MB state
    float*  M   = (float*)((char*)d_ws + sizeof(float2) * (size_t)NSTATE);
    float*  out = (float*)d_out;

    init_state_kernel<<<NSTATE / 256, 256, 0, stream>>>(psi);
    build_matrix_kernel<<<1, 32, 0, stream>>>(params, M);
    for (int q = 0; q < 19; ++q)
        apply_gate_kernel<<<NGROUP / TPB, TPB, 0, stream>>>(psi, M, q);
    probs_kernel<<<NSTATE / 256, 256, 0, stream>>>(psi, out);
}